// GNNModel_44195213476075
// MI455X (gfx1250) — compile-verified
//
#include <hip/hip_runtime.h>
#include <hip/hip_bf16.h>

// ---------------------------------------------------------------------------
// GCN (3x GCNConv + mean pool) for MI455X / gfx1250.
// - GEMMs: V_WMMA_F32_16X16X4_F32 (exact f32 matrix path, CDNA5 wave32 WMMA)
// - A tile staged to LDS via GLOBAL_LOAD_ASYNC_TO_LDS_B128 (ASYNCcnt DMA path)
// - Edge aggregation: L2-resident f32 atomics (tables ~5MB << 192MB L2)
// ---------------------------------------------------------------------------

typedef __attribute__((ext_vector_type(2))) float v2f;
typedef __attribute__((ext_vector_type(8))) float v8f;

#define GCN_N 10000
#define GCN_H 128
#define GCN_G 64

// ---- degree / normalization ------------------------------------------------
__global__ void deg_init_kernel(float* __restrict__ dis, int n) {
    int i = blockIdx.x * blockDim.x + threadIdx.x;
    if (i < n) dis[i] = 1.0f;  // +1 for self loop
}

__global__ void deg_accum_kernel(const int* __restrict__ dst,
                                 float* __restrict__ dis, int e) {
    int i = blockIdx.x * blockDim.x + threadIdx.x;
    if (i < e) atomicAdd(&dis[dst[i]], 1.0f);
}

__global__ void deg_finish_kernel(float* __restrict__ dis, int n) {
    int i = blockIdx.x * blockDim.x + threadIdx.x;
    if (i < n) dis[i] = 1.0f / sqrtf(dis[i]);  // deg >= 1
}

// ---- dense GEMM: out[N,128] = relu?(in)[N,128] @ W[128,128] ----------------
// Block: 256 threads = 8 wave32s. Block -> 16-row tile; wave w -> 16-col tile.
// A tile DMA'd to LDS (async-to-LDS, ASYNCcnt), padded stride 132 floats so
// the per-lane A-fragment reads are conflict-free ds_load_b64.
template <bool RELU>
__global__ void __launch_bounds__(256)
gemm_wmma_f32_kernel(const float* __restrict__ in, const float* __restrict__ W,
                     float* __restrict__ out) {
    __shared__ float ldsA[16 * 132];
    const int m0  = blockIdx.x * 16;
    const int tid = threadIdx.x;

    // Pull the whole 64KB weight matrix toward the caches while we stage A.
    __builtin_prefetch(W + tid * 64, 0, 1);  // 256 thr * 256B = 64KB

    // Async DMA of the 16x128 f32 A tile into LDS: 512 x 16B chunks,
    // 2 chunks per thread. Row r -> LDS offset r*528 (132-float pad).
    {
        const uint32_t ldsBase = (uint32_t)(uintptr_t)&ldsA[0];
        for (int c = tid; c < 512; c += 256) {
            const int r  = c >> 5;        // row in tile (0..15)
            const int cc = c & 31;        // 16B chunk in row (0..31)
            const uint32_t loff = ldsBase + (uint32_t)(r * 528 + cc * 16);
            const float* gp = in + (size_t)(m0 + r) * GCN_H + cc * 4;
            asm volatile("global_load_async_to_lds_b128 %0, %1, off"
                         :
                         : "v"(loff), "v"(gp)
                         : "memory");
        }
    }
#if __has_builtin(__builtin_amdgcn_s_wait_asynccnt)
    __builtin_amdgcn_s_wait_asynccnt(0);
#else
    asm volatile("s_wait_asynccnt 0" ::: "memory");
#endif
    __syncthreads();

    const int lane = tid & 31;
    const int wave = tid >> 5;        // 0..7 -> output col tile
    const int half = lane >> 4;       // 0/1: selects K pair (A) / K rows (B)
    const int l16  = lane & 15;       // A: row M,  B/D: col N
    const int n    = wave * 16 + l16;

    v8f acc = {};
    const float* arow = ldsA + l16 * 132;
#pragma unroll
    for (int k = 0; k < 128; k += 4) {
        const int kk = k + 2 * half;
        v2f a;
        a.x = arow[kk];
        a.y = arow[kk + 1];
        if (RELU) {  // fused ReLU of the previous layer's activations
            a.x = fmaxf(a.x, 0.0f);
            a.y = fmaxf(a.y, 0.0f);
        }
        v2f b;
        b.x = W[(size_t)kk * GCN_H + n];
        b.y = W[(size_t)(kk + 1) * GCN_H + n];
        // D = A(16x4,f32) x B(4x16,f32) + C
        acc = __builtin_amdgcn_wmma_f32_16x16x4_f32(
            false, a, false, b, (short)0, acc, false, false);
    }

    // D layout: VGPR j -> rows M=j (lanes 0-15) and M=j+8 (lanes 16-31)
#pragma unroll
    for (int j = 0; j < 8; ++j) {
        int row = m0 + j + half * 8;
        out[(size_t)row * GCN_H + n] = acc[j];
    }
}

// ---- self-loop + bias: out[i,c] = h[i,c]*dis[i]^2 + b[c] -------------------
__global__ void self_bias_kernel(const float* __restrict__ h,
                                 const float* __restrict__ dis,
                                 const float* __restrict__ b,
                                 float* __restrict__ out, int total) {
    int i = blockIdx.x * blockDim.x + threadIdx.x;
    if (i >= total) return;
    int node = i >> 7, c = i & 127;
    float d = dis[node];
    out[i] = h[i] * d * d + b[c];
}

// ---- edge scatter: out[dst] += h[src] * dis[src]*dis[dst] ------------------
// One wave per edge; each lane handles a float4 (32 lanes x 4 = 128 feats).
__global__ void edge_scatter_kernel(const float* __restrict__ h,
                                    const int* __restrict__ src,
                                    const int* __restrict__ dst,
                                    const float* __restrict__ dis,
                                    float* __restrict__ out, int e) {
    int wave = (blockIdx.x * blockDim.x + threadIdx.x) >> 5;
    int lane = threadIdx.x & 31;
    if (wave >= e) return;
    int s = src[wave];
    int d = dst[wave];
    float coef = dis[s] * dis[d];
    const float4* hp = (const float4*)(h + (size_t)s * GCN_H);
    float4 v = hp[lane];
    float* op = out + (size_t)d * GCN_H + lane * 4;
    atomicAdd(op + 0, v.x * coef);
    atomicAdd(op + 1, v.y * coef);
    atomicAdd(op + 2, v.z * coef);
    atomicAdd(op + 3, v.w * coef);
}

// ---- mean pool -------------------------------------------------------------
__global__ void pool_zero_kernel(float* __restrict__ p, int total) {
    int i = blockIdx.x * blockDim.x + threadIdx.x;
    if (i < total) p[i] = 0.0f;
}

// One wave per node: lane -> float4 of features; lane 0 bumps the count.
__global__ void pool_accum_kernel(const float* __restrict__ h,
                                  const int* __restrict__ batch,
                                  float* __restrict__ pool,
                                  float* __restrict__ cnt, int n) {
    int wave = (blockIdx.x * blockDim.x + threadIdx.x) >> 5;
    int lane = threadIdx.x & 31;
    if (wave >= n) return;
    int g = batch[wave];
    const float4* hp = (const float4*)(h + (size_t)wave * GCN_H);
    float4 v = hp[lane];
    float* op = pool + (size_t)g * GCN_H + lane * 4;
    atomicAdd(op + 0, v.x);
    atomicAdd(op + 1, v.y);
    atomicAdd(op + 2, v.z);
    atomicAdd(op + 3, v.w);
    if (lane == 0) atomicAdd(&cnt[g], 1.0f);
}

__global__ void pool_div_kernel(const float* __restrict__ pool,
                                const float* __restrict__ cnt,
                                float* __restrict__ out, int total) {
    int i = blockIdx.x * blockDim.x + threadIdx.x;
    if (i >= total) return;
    int g = i >> 7;
    out[i] = pool[i] / fmaxf(cnt[g], 1.0f);
}

// ---------------------------------------------------------------------------
extern "C" void kernel_launch(void* const* d_in, const int* in_sizes, int n_in,
                              void* d_out, int out_size, void* d_ws, size_t ws_size,
                              hipStream_t stream) {
    const float* x     = (const float*)d_in[0];
    const int*   ei    = (const int*)d_in[1];
    const int*   batch = (const int*)d_in[2];
    const float* W0    = (const float*)d_in[3];
    const float* b0    = (const float*)d_in[4];
    const float* W1    = (const float*)d_in[5];
    const float* b1    = (const float*)d_in[6];
    const float* W2    = (const float*)d_in[7];
    const float* b2    = (const float*)d_in[8];
    float* out = (float*)d_out;

    const int N = GCN_N;
    const int E = in_sizes[1] / 2;   // edge_index is [2, E]
    const int H = GCN_H;
    const int G = GCN_G;
    const int* src = ei;
    const int* dst = ei + E;

    // workspace layout (floats): dis[N] | bufA[N*H] | bufB[N*H] | pool[G*H] | cnt[G]
    float* ws   = (float*)d_ws;
    float* dis  = ws;
    float* bufA = ws + N;                       // GEMM output h
    float* bufB = bufA + (size_t)N * H;         // aggregated output / next input
    float* pool = bufB + (size_t)N * H;
    float* cnt  = pool + (size_t)G * H;

    const int T = 256;

    // 1) symmetric-normalization coefficients
    deg_init_kernel<<<(N + T - 1) / T, T, 0, stream>>>(dis, N);
    deg_accum_kernel<<<(E + T - 1) / T, T, 0, stream>>>(dst, dis, E);
    deg_finish_kernel<<<(N + T - 1) / T, T, 0, stream>>>(dis, N);

    const float* Ws[3] = {W0, W1, W2};
    const float* bs[3] = {b0, b1, b2};
    const int gemm_blocks    = N / 16;                 // 10000/16 = 625 exact
    const int scatter_blocks = (E * 32 + T - 1) / T;   // one wave per edge
    const int nh_blocks      = (N * H + T - 1) / T;

    const float* layer_in = x;
    for (int l = 0; l < 3; ++l) {
        if (l == 0)
            gemm_wmma_f32_kernel<false><<<gemm_blocks, T, 0, stream>>>(layer_in, Ws[l], bufA);
        else
            gemm_wmma_f32_kernel<true><<<gemm_blocks, T, 0, stream>>>(layer_in, Ws[l], bufA);
        self_bias_kernel<<<nh_blocks, T, 0, stream>>>(bufA, dis, bs[l], bufB, N * H);
        edge_scatter_kernel<<<scatter_blocks, T, 0, stream>>>(bufA, src, dst, dis, bufB, E);
        layer_in = bufB;
    }

    // 2) global mean pool (pool and cnt are contiguous -> one zeroing pass)
    pool_zero_kernel<<<(G * H + G + T - 1) / T, T, 0, stream>>>(pool, G * H + G);
    pool_accum_kernel<<<(N * 32 + T - 1) / T, T, 0, stream>>>(bufB, batch, pool, cnt, N);
    pool_div_kernel<<<(G * H + T - 1) / T, T, 0, stream>>>(pool, cnt, out, G * H);
}